// FixedSpatioTemporalGraphBuilder_51238959841299
// MI455X (gfx1250) — compile-verified
//
#include <hip/hip_runtime.h>
#include <hip/hip_bf16.h>

typedef float v2f __attribute__((ext_vector_type(2)));
typedef float v8f __attribute__((ext_vector_type(8)));
typedef unsigned int v4u __attribute__((ext_vector_type(4)));
typedef unsigned int v8u __attribute__((ext_vector_type(8)));

#define GS        32
#define GH        7
#define GW        7
#define NPF       49              // nodes per frame
#define T_FRAMES  64
#define B_BATCH   8
#define C_CH      3
#define HW        224
#define PATCH     3072            // C*GS*GS
#define SORT_N    4096
#define NTHREADS  256
#define NODES_PER_B (T_FRAMES*NPF)        // 3136
#define X_SIZE    (B_BATCH*NODES_PER_B*10) // 250880
#define E_TOTAL   32218
#define F_EPS     1e-8f

__device__ __forceinline__ int tcnt(int t) {
    int c = 0;
    if (t + 1 < T_FRAMES) c++;
    if (t - 1 >= 0)       c++;
    if (t + 2 < T_FRAMES) c++;
    if (t - 2 >= 0)       c++;
    return c;
}
__device__ __forceinline__ int scnt(int i, int j) {
    int ci = 1 + (i > 0) + (i < GH - 1);
    int cj = 1 + (j > 0) + (j < GW - 1);
    return ci * cj - 1;
}

__global__ __launch_bounds__(NTHREADS)
void patch_stats_kernel(const float* __restrict__ video, float* __restrict__ out)
{
    __shared__ float sdata[SORT_N];
    __shared__ float sacc[4];

    const int blk = blockIdx.x;
    const int b   = blk / NODES_PER_B;
    const int r   = blk % NODES_PER_B;
    const int t   = r / NPF;
    const int pos = r % NPF;
    const int gi  = pos / GW;
    const int gj  = pos % GW;

    const int tid = threadIdx.x;
    if (tid < 4) sacc[tid] = 0.0f;

    // ---- pad sort slots with +inf (overlaps the TDM DMA; disjoint LDS range) ----
    for (int e = PATCH + tid; e < SORT_N; e += NTHREADS)
        sdata[e] = __builtin_inff();

    // ---- Tensor Data Mover: DMA the whole 3D patch tile into LDS ----
    // tile: x = 32 contiguous floats, y = 32 rows @ stride 224,
    //       z = 3 channels @ stride T*H*W = 3,211,264 elements.
    // TDM LDS fill order (x,y,z contiguous) == reference channel-major flatten.
    if (tid < 32) {   // wave 0 only (uniform branch; TDM ignores EXEC)
        const long long row0  = (((long long)(b * C_CH) * T_FRAMES + t) * HW
                                 + (gi * GS));
        const long long elem0 = row0 * HW + (gj * GS);
        const unsigned long long gaddr =
            (unsigned long long)(const void*)(video + elem0);
        const unsigned int ldsa =
            (unsigned int)(unsigned long long)(void*)&sdata[0];

        // D# group 0: count=1 | lds_addr | global_addr[56:0] | type=2
        const v4u g0 = { 1u,
                         ldsa,
                         (unsigned int)gaddr,
                         (unsigned int)((gaddr >> 32) & 0x1FFFFFFu) | (2u << 30) };
        // D# group 1: data_size=4B; tensor_dim0=32, tensor_dim1=32;
        // tile_dim0=32, tile_dim1=32, tile_dim2=3;
        // tensor_dim0_stride=224, tensor_dim1_stride=3211264 (=0x310000)
        const v8u g1 = { 0x00020000u,   // data_size=2 (4 bytes)
                         0x00200000u,   // abar=0 | tensor_dim0.lo16=32
                         0x00200000u,   // tensor_dim0.hi16=0 | tensor_dim1.lo16=32
                         0x00200000u,   // tensor_dim1.hi16=0 | tile_dim0=32
                         0x00030020u,   // tile_dim1=32 | tile_dim2=3
                         224u,          // tensor_dim0_stride[31:0]
                         0u,            // stride0[47:32]=0 | stride1[15:0]=0
                         49u };         // tensor_dim1_stride[47:16] = 0x31
        // D# group 2: tensor_dim2=3; rest unused
        const v4u g2 = { 3u, 0u, 0u, 0u };
        const v4u g3 = { 0u, 0u, 0u, 0u };

        asm volatile("tensor_load_to_lds %0, %1, %2, %3"
                     :: "s"(g0), "s"(g1), "s"(g2), "s"(g3)
                     : "memory");
        __builtin_amdgcn_s_wait_tensorcnt(0);
    }
    __syncthreads();

    // ---- per-wave WMMA moment reduction: Sum x, x^2, x^3, x^4 ----
    // A (16x4 f32): lanes 0-15 hold K=0,1 ; lanes 16-31 hold K=2,3; row M = lane%16
    // B = all-ones 4x16  ->  D[m][n] = C[m][n] + sum_k A[m][k]
    const int wave = tid >> 5;
    const int lane = tid & 31;
    const v2f ones = {1.0f, 1.0f};
    v8f acc1 = {}, acc2 = {}, acc3 = {}, acc4 = {};
    const int seg     = wave * (PATCH / 8);                 // 384 floats per wave
    const int laneoff = (lane & 15) * 4 + (lane >> 4) * 2;  // even -> 8B aligned ds_load_b64
    for (int it = 0; it < 6; ++it) {
        const int cb = seg + it * 64 + laneoff;
        v2f a  = *(const v2f*)(&sdata[cb]);
        v2f a2 = a * a;
        v2f a3 = a2 * a;
        v2f a4 = a2 * a2;
        acc1 = __builtin_amdgcn_wmma_f32_16x16x4_f32(false, a,  false, ones, (short)0, acc1, false, false);
        acc2 = __builtin_amdgcn_wmma_f32_16x16x4_f32(false, a2, false, ones, (short)0, acc2, false, false);
        acc3 = __builtin_amdgcn_wmma_f32_16x16x4_f32(false, a3, false, ones, (short)0, acc3, false, false);
        acc4 = __builtin_amdgcn_wmma_f32_16x16x4_f32(false, a4, false, ones, (short)0, acc4, false, false);
    }
    // fold C/D layout: 8 VGPRs hold rows m=0..7 (lanes 0-15) / m=8..15 (lanes 16-31)
    float s1 = 0.f, s2 = 0.f, s3 = 0.f, s4 = 0.f;
    for (int rr = 0; rr < 8; ++rr) { s1 += acc1[rr]; s2 += acc2[rr]; s3 += acc3[rr]; s4 += acc4[rr]; }
    s1 += __shfl_xor(s1, 16, 32);
    s2 += __shfl_xor(s2, 16, 32);
    s3 += __shfl_xor(s3, 16, 32);
    s4 += __shfl_xor(s4, 16, 32);
    if (lane == 0) {                 // ds_add_f32 atomics across the 8 waves
        atomicAdd(&sacc[0], s1);
        atomicAdd(&sacc[1], s2);
        atomicAdd(&sacc[2], s3);
        atomicAdd(&sacc[3], s4);
    }
    __syncthreads();

    // ---- bitonic sort (ascending) of 4096 slots in LDS ----
    for (int k = 2; k <= SORT_N; k <<= 1) {
        for (int j = k >> 1; j > 0; j >>= 1) {
            for (int i = tid; i < SORT_N; i += NTHREADS) {
                const int ij = i ^ j;
                if (ij > i) {
                    const float a  = sdata[i];
                    const float bb = sdata[ij];
                    const bool up  = ((i & k) == 0);
                    if ((a > bb) == up) { sdata[i] = bb; sdata[ij] = a; }
                }
            }
            __syncthreads();
        }
    }

    if (tid == 0) {
        const float n  = (float)PATCH;
        const float S1 = sacc[0], S2 = sacc[1], S3 = sacc[2], S4 = sacc[3];
        const float mean = S1 / n;
        float var = (S2 - n * mean * mean) / (n - 1.0f);
        if (var < 0.0f) var = 0.0f;
        const float stdv   = sqrtf(var);
        const float mn     = sdata[0];
        const float mx     = sdata[PATCH - 1];
        const float energy = S2 / n;
        const float rms    = sqrtf(energy + F_EPS);
        // central moments from raw moments
        const float M2 = S2 / n, M3 = S3 / n, M4 = S4 / n;
        const float m3c = M3 - 3.0f * mean * M2 + 2.0f * mean * mean * mean;
        const float m4c = M4 - 4.0f * mean * M3 + 6.0f * mean * mean * M2
                          - 3.0f * mean * mean * mean * mean;
        const float d    = stdv + F_EPS;
        const float skew = m3c / (d * d * d);
        const float kurt = m4c / (d * d * d * d) - 3.0f;
        const float median = sdata[(PATCH - 1) / 2];                 // 1535 (lower median)
        const float q25 = 0.25f * sdata[767]  + 0.75f * sdata[768];  // pos 767.75
        const float q75 = 0.75f * sdata[2303] + 0.25f * sdata[2304]; // pos 2303.25
        float* o = out + (size_t)blk * 10;
        o[0] = mean;  o[1] = stdv; o[2] = mn;     o[3] = mx;  o[4] = energy;
        o[5] = rms;   o[6] = skew; o[7] = kurt;   o[8] = median; o[9] = q75 - q25;
    }
}

__global__ __launch_bounds__(256)
void edge_kernel(float* __restrict__ out)
{
    const int node = blockIdx.x * blockDim.x + threadIdx.x;
    if (node >= T_FRAMES * NPF) return;
    const int t   = node / NPF;
    const int pos = node % NPF;
    const int gi  = pos / GW, gj = pos % GW;

    // closed-form emission offset: frames before t, then nodes before pos in frame t
    long long off = 0;
    for (int tt = 0; tt < t; ++tt) off += 312 + NPF * tcnt(tt);
    const int tc = tcnt(t);
    for (int p = 0; p < pos; ++p) off += scnt(p / GW, p % GW) + tc;

    float* src = out + X_SIZE;
    float* dst = out + X_SIZE + E_TOTAL;
    long long w = off;
    // spatial 8-neighborhood, reference emission order
    for (int di = -1; di <= 1; ++di)
        for (int dj = -1; dj <= 1; ++dj) {
            if (di == 0 && dj == 0) continue;
            const int ni = gi + di, nj = gj + dj;
            if (ni >= 0 && ni < GH && nj >= 0 && nj < GW) {
                src[w] = (float)node;
                dst[w] = (float)(t * NPF + ni * GW + nj);
                ++w;
            }
        }
    // temporal order: +1, -1, +2, -2
    const int dts[4] = {1, -1, 2, -2};
    for (int q = 0; q < 4; ++q) {
        const int tt = t + dts[q];
        if (tt >= 0 && tt < T_FRAMES) {
            src[w] = (float)node;
            dst[w] = (float)(tt * NPF + pos);
            ++w;
        }
    }
}

extern "C" void kernel_launch(void* const* d_in, const int* in_sizes, int n_in,
                              void* d_out, int out_size, void* d_ws, size_t ws_size,
                              hipStream_t stream) {
    (void)in_sizes; (void)n_in; (void)d_ws; (void)ws_size; (void)out_size;
    const float* video = (const float*)d_in[0];
    float* out = (float*)d_out;
    patch_stats_kernel<<<B_BATCH * NODES_PER_B, NTHREADS, 0, stream>>>(video, out);
    edge_kernel<<<(T_FRAMES * NPF + 255) / 256, 256, 0, stream>>>(out);
}